// MeshModel_70093866270810
// MI455X (gfx1250) — compile-verified
//
#include <hip/hip_runtime.h>
#include <hip/hip_bf16.h>

#define EPSF 1e-6f

typedef float v2f __attribute__((ext_vector_type(2)));
typedef float v8f __attribute__((ext_vector_type(8)));
typedef float f4  __attribute__((ext_vector_type(4)));

// ---------------------------------------------------------------------------
// Wave-level sum reduction using V_WMMA_F32_16X16X4_F32 (full f32 precision).
// A (16x4 f32): lane L (0..15) supplies row M=L at K=0 (a.x) / K=1 (a.y=0);
// lane L (16..31) supplies row M=L-16 at K=2 / K=3. With B = all-ones,
// D[m][n] = acc[m] + acc[m+16] (broadcast over n). Per-lane sum of the 8
// accumulator VGPRs gives sum(D[0..7][*]) on the low half and
// sum(D[8..15][*]) on the high half; one xor-16 shuffle completes the wave sum.
// ---------------------------------------------------------------------------
__device__ __forceinline__ float wave_sum32(float x) {
#if __has_builtin(__builtin_amdgcn_wmma_f32_16x16x4_f32)
  v2f a;  a.x = x;   a.y = 0.0f;
  v2f bm; bm.x = 1.0f; bm.y = 1.0f;
  v8f c = {};
  c = __builtin_amdgcn_wmma_f32_16x16x4_f32(
      /*neg_a=*/false, a, /*neg_b=*/false, bm,
      /*c_mod=*/(short)0, c, /*reuse_a=*/false, /*reuse_b=*/false);
  float s = c[0] + c[1] + c[2] + c[3] + c[4] + c[5] + c[6] + c[7];
  return s + __shfl_xor(s, 16, 32);
#else
  for (int off = 16; off > 0; off >>= 1) x += __shfl_xor(x, off, 32);
  return x;
#endif
}

// ---------------------------------------------------------------------------
// K0: zero the neighbor-accumulator workspace and the two loss slots.
// ---------------------------------------------------------------------------
__global__ void k_zero(float* __restrict__ nbr, long n, float* __restrict__ losses) {
  long stride = (long)gridDim.x * blockDim.x;
  long t0 = (long)blockIdx.x * blockDim.x + threadIdx.x;
  for (long t = t0; t < n; t += stride) nbr[t] = 0.0f;
  if (t0 < 2) losses[t0] = 0.0f;
}

// ---------------------------------------------------------------------------
// K1: v = f(template, displace, center)  ->  ws_v  (B, nv, 3)
// ---------------------------------------------------------------------------
__global__ void k_compute_v(const float* __restrict__ disp,
                            const float* __restrict__ center,
                            const float* __restrict__ tv,
                            float* __restrict__ v,
                            int B, long nv) {
  long total  = (long)B * nv;
  long stride = (long)gridDim.x * blockDim.x;
  for (long t = (long)blockIdx.x * blockDim.x + threadIdx.x; t < total; t += stride) {
    int  b = (int)(t / nv);
    long i = t % nv;
#pragma unroll
    for (int c = 0; c < 3; ++c) {
      float tvc  = tv[i * 3 + c];
      float a    = fabsf(tvc);
      float base = __logf(a / (1.0f - a));              // logit(|tv|)
      float d    = disp[((long)b * nv + i) * 3 + c];
      float sig  = 1.0f / (1.0f + __expf(-(base + d))); // sigmoid
      float sgn  = (tvc > 0.0f) ? 1.0f : ((tvc < 0.0f) ? -1.0f : 0.0f);
      float vv   = sig * sgn;
      float cen  = tanhf(center[b * 3 + c]);
      float rp   = vv > 0.0f ? vv : 0.0f;
      float rn   = vv < 0.0f ? -vv : 0.0f;
      vv = rp * (1.0f - cen) - rn * (cen + 1.0f) + cen;
      v[((long)b * nv + i) * 3 + c] = vv;
    }
  }
}

// ---------------------------------------------------------------------------
// K2: replicate each batch row (R floats) NV times, b128 vectorized,
// non-temporal stores (outputs are write-once, never re-read).
// out[(b*NV + k)*R + r] = src[b*R + r]
// ---------------------------------------------------------------------------
__global__ void k_replicate4(const f4* __restrict__ src, f4* __restrict__ dst,
                             int NV, long R4, long total) {
  long stride = (long)gridDim.x * blockDim.x;
  for (long t = (long)blockIdx.x * blockDim.x + threadIdx.x; t < total; t += stride) {
    long ob = t / R4;
    long r  = t - ob * R4;
    long b  = ob / NV;
    f4 val = src[b * R4 + r];
    __builtin_nontemporal_store(val, &dst[ob * R4 + r]);
  }
}

__global__ void k_replicate1(const float* __restrict__ src, float* __restrict__ dst,
                             int NV, long R, long total) {
  long stride = (long)gridDim.x * blockDim.x;
  for (long t = (long)blockIdx.x * blockDim.x + threadIdx.x; t < total; t += stride) {
    long ob = t / R;
    long r  = t - ob * R;
    long b  = ob / NV;
    __builtin_nontemporal_store(src[b * R + r], &dst[ob * R + r]);
  }
}

// ---------------------------------------------------------------------------
// K3: segment-sum scatter: nbr[b, dst[e], :] += v[b, src[e], :]
// ---------------------------------------------------------------------------
__global__ void k_scatter(const int* __restrict__ esrc, const int* __restrict__ edst,
                          const float* __restrict__ v, float* __restrict__ nbr,
                          long nE, long nv, long total) {
  long stride = (long)gridDim.x * blockDim.x;
  for (long t = (long)blockIdx.x * blockDim.x + threadIdx.x; t < total; t += stride) {
    long e = t % nE;
    long b = t / nE;
    int s = esrc[e];
    int d = edst[e];
    const float* vp = v   + (b * nv + s) * 3;
    float*       np = nbr + (b * nv + d) * 3;
    __builtin_prefetch(vp, 0, 0);
#pragma unroll
    for (int c = 0; c < 3; ++c) atomicAdd(&np[c], vp[c]);
  }
}

// ---------------------------------------------------------------------------
// K4: lap_loss = (1/B) * sum_{b,i,c} (v - nbr/deg)^2 ; WMMA wave reduction.
// ---------------------------------------------------------------------------
__global__ void k_laploss(const float* __restrict__ v, const float* __restrict__ nbr,
                          const int* __restrict__ deg, float* __restrict__ out,
                          long nv, long total, float scale) {
  float acc = 0.0f;
  long stride = (long)gridDim.x * blockDim.x;
  for (long t = (long)blockIdx.x * blockDim.x + threadIdx.x; t < total; t += stride) {
    long i = t % nv;
    float inv_deg = 1.0f / (float)deg[i];
#pragma unroll
    for (int c = 0; c < 3; ++c) {
      float lx = v[t * 3 + c] - nbr[t * 3 + c] * inv_deg;
      acc = fmaf(lx, lx, acc);
    }
  }
  float tot = wave_sum32(acc);   // EXEC is all-ones here (uniform control flow)
  if ((threadIdx.x & 31) == 0) atomicAdd(out, tot * scale);
}

// ---------------------------------------------------------------------------
// K5: flat_loss = (1/B) * sum_{b,q} (cos+1)^2  ; WMMA wave reduction.
// ---------------------------------------------------------------------------
__global__ void k_flatloss(const float* __restrict__ v,
                           const int* __restrict__ v0, const int* __restrict__ v1,
                           const int* __restrict__ v2, const int* __restrict__ v3,
                           float* __restrict__ out,
                           long nv, long E2, long total, float scale) {
  float acc = 0.0f;
  long stride = (long)gridDim.x * blockDim.x;
  for (long t = (long)blockIdx.x * blockDim.x + threadIdx.x; t < total; t += stride) {
    long q = t % E2;
    long b = t / E2;
    const float* vb = v + b * nv * 3;
    long i0 = v0[q], i1 = v1[q], i2 = v2[q], i3 = v3[q];
    float p0[3], a1[3], b1[3], b2[3];
#pragma unroll
    for (int c = 0; c < 3; ++c) {
      p0[c] = vb[i0 * 3 + c];
      a1[c] = vb[i1 * 3 + c] - p0[c];
      b1[c] = vb[i2 * 3 + c] - p0[c];
      b2[c] = vb[i3 * 3 + c] - p0[c];
    }
    float a1l2 = a1[0]*a1[0] + a1[1]*a1[1] + a1[2]*a1[2];
    float a1l1 = sqrtf(a1l2 + EPSF);
    float inv_a1l2 = 1.0f / (a1l2 + EPSF);

    // component 1 (p2 - p0)
    float bl2  = b1[0]*b1[0] + b1[1]*b1[1] + b1[2]*b1[2];
    float bl1  = sqrtf(bl2 + EPSF);
    float ab   = a1[0]*b1[0] + a1[1]*b1[1] + a1[2]*b1[2];
    float cos1 = ab / (a1l1 * bl1 + EPSF);
    float sin1 = sqrtf(1.0f - cos1 * cos1 + EPSF);
    float k1   = ab * inv_a1l2;
    float cb1[3];
#pragma unroll
    for (int c = 0; c < 3; ++c) cb1[c] = b1[c] - a1[c] * k1;
    float l1 = bl1 * sin1;

    // component 2 (p3 - p0)
    bl2  = b2[0]*b2[0] + b2[1]*b2[1] + b2[2]*b2[2];
    bl1  = sqrtf(bl2 + EPSF);
    ab   = a1[0]*b2[0] + a1[1]*b2[1] + a1[2]*b2[2];
    float cos2 = ab / (a1l1 * bl1 + EPSF);
    float sin2 = sqrtf(1.0f - cos2 * cos2 + EPSF);
    float k2   = ab * inv_a1l2;
    float cb2[3];
#pragma unroll
    for (int c = 0; c < 3; ++c) cb2[c] = b2[c] - a1[c] * k2;
    float l2 = bl1 * sin2;

    float cosv = (cb1[0]*cb2[0] + cb1[1]*cb2[1] + cb1[2]*cb2[2]) / (l1 * l2 + EPSF);
    float u = cosv + 1.0f;
    acc = fmaf(u, u, acc);
  }
  float tot = wave_sum32(acc);
  if ((threadIdx.x & 31) == 0) atomicAdd(out, tot * scale);
}

// ---------------------------------------------------------------------------
static inline int grid_for(long n, int threads, int cap = 16384) {
  long b = (n + threads - 1) / threads;
  if (b < 1) b = 1;
  if (b > cap) b = cap;
  return (int)b;
}

extern "C" void kernel_launch(void* const* d_in, const int* in_sizes, int n_in,
                              void* d_out, int out_size, void* d_ws, size_t ws_size,
                              hipStream_t stream) {
  // Inputs (setup_inputs order):
  // 0 displace (B,nv,3) f32 | 1 center (B,1,3) f32 | 2 texture (B,nv,3) f32
  // 3 template_vertices (1,nv,3) f32 | 4 lap_src i32 | 5 lap_dst i32
  // 6 deg i32 | 7 v0s | 8 v1s | 9 v2s | 10 v3s | 11 num_views (scalar)
  const float* disp   = (const float*)d_in[0];
  const float* center = (const float*)d_in[1];
  const float* tex    = (const float*)d_in[2];
  const float* tv     = (const float*)d_in[3];
  const int*   lsrc   = (const int*)d_in[4];
  const int*   ldst   = (const int*)d_in[5];
  const int*   deg    = (const int*)d_in[6];
  const int*   v0s    = (const int*)d_in[7];
  const int*   v1s    = (const int*)d_in[8];
  const int*   v2s    = (const int*)d_in[9];
  const int*   v3s    = (const int*)d_in[10];

  const long nv = in_sizes[3] / 3;            // 262144
  const int  B  = in_sizes[0] / in_sizes[3];  // 4
  const long R  = nv * 3;                     // floats per batch row
  const int  NV = (int)(((long)out_size - 2) / (2L * B * R)); // num_views
  const long nE = in_sizes[4];                // directed edges
  const long E2 = in_sizes[7];                // interior edges

  float* ws_v   = (float*)d_ws;               // B*R floats
  float* ws_nbr = ws_v + (size_t)B * R;       // B*R floats

  float* verts  = (float*)d_out;                         // NV*B*R
  float* tex_o  = verts + (size_t)NV * B * R;            // NV*B*R
  float* losses = tex_o + (size_t)NV * B * R;            // [lap, flat]

  const int T = 256;
  const float invB = 1.0f / (float)B;

  // K0: zero neighbor accumulator + loss slots
  k_zero<<<grid_for(B * R, T), T, 0, stream>>>(ws_nbr, B * R, losses);

  // K1: vertex transform -> ws_v
  k_compute_v<<<grid_for((long)B * nv, T), T, 0, stream>>>(disp, center, tv, ws_v, B, nv);

  // K2: replicate v -> verts_out and texture -> tex_out (NT b128 stores)
  if ((R & 3) == 0) {
    long R4 = R >> 2;
    long tot = (long)B * NV * R4;
    k_replicate4<<<grid_for(tot, T), T, 0, stream>>>((const f4*)ws_v, (f4*)verts, NV, R4, tot);
    k_replicate4<<<grid_for(tot, T), T, 0, stream>>>((const f4*)tex, (f4*)tex_o, NV, R4, tot);
  } else {
    long tot = (long)B * NV * R;
    k_replicate1<<<grid_for(tot, T), T, 0, stream>>>(ws_v, verts, NV, R, tot);
    k_replicate1<<<grid_for(tot, T), T, 0, stream>>>(tex, tex_o, NV, R, tot);
  }

  // K3: Laplacian neighbor scatter-sum
  {
    long tot = nE * B;
    k_scatter<<<grid_for(tot, T), T, 0, stream>>>(lsrc, ldst, ws_v, ws_nbr, nE, nv, tot);
  }

  // K4: Laplacian loss (WMMA wave reduction)
  {
    long tot = (long)B * nv;
    k_laploss<<<grid_for(tot, T), T, 0, stream>>>(ws_v, ws_nbr, deg, &losses[0], nv, tot, invB);
  }

  // K5: flatness loss (WMMA wave reduction)
  {
    long tot = (long)B * E2;
    k_flatloss<<<grid_for(tot, T), T, 0, stream>>>(ws_v, v0s, v1s, v2s, v3s, &losses[1],
                                                   nv, E2, tot, invB);
  }
}